// EnhancedHeteroGNN_86629490360459
// MI455X (gfx1250) — compile-verified
//
#include <hip/hip_runtime.h>
#include <math.h>

#define NN   50000
#define EE   600000
#define RRr  64
#define VVv  (NN - RRr)
#define NFt  9
#define HDt  128
#define Ht   4
#define C1t  32
#define EDt  8
#define INt  41
#define KPAD 44

typedef float v2f __attribute__((ext_vector_type(2)));
typedef float v8f __attribute__((ext_vector_type(8)));

__device__ __forceinline__ void atomicMaxF(float* addr, float val) {
    unsigned int* ua = (unsigned int*)addr;
    unsigned int old = *ua;
    while (__uint_as_float(old) < val) {
        unsigned int assumed = old;
        old = atomicCAS(ua, assumed, __float_as_uint(val));
        if (old == assumed) break;
    }
}

// ---------------------------------------------------------------------------
// fp32 WMMA GEMM: C[M x Ncols] = A[M x K] * B[K x Ncols], optional bias+relu.
// M = gridDim.x*16, blockDim = 128 (4 waves). B must have full K rows (padded).
// Inner loop: ds_load_2addr (A from LDS) + 2 coalesced global loads (B) + V_WMMA.
// ---------------------------------------------------------------------------
__global__ __launch_bounds__(128)
void k_wmma_gemm(const float* __restrict__ A, const float* __restrict__ B,
                 float* __restrict__ C, const float* __restrict__ bias,
                 int K, int Ncols, int act)
{
    __shared__ float As[16 * 256];
    const int tid  = threadIdx.x;
    const int row0 = blockIdx.x * 16;

    for (int idx = tid; idx < 16 * K; idx += 128) {
        int r = idx / K, c = idx - r * K;
        As[idx] = A[(size_t)(row0 + r) * K + c];
    }
    __syncthreads();

    const int lane = tid & 31;
    const int wv   = tid >> 5;
    const int lo   = lane & 15;
    const int hi   = lane >> 4;

    for (int nt = wv; nt * 16 < Ncols; nt += 4) {
        const int col = nt * 16 + lo;
        // f32 WMMA operand layout (ISA 7.12.2): lane-half selects K pair,
        // VGPR index selects K within pair, for both A (lane=M) and B (lane=N).
        const float* ap = As + lo * K + 2 * hi;
        const float* bp = B + (size_t)(2 * hi) * Ncols + col;
        v8f acc = {0.f, 0.f, 0.f, 0.f, 0.f, 0.f, 0.f, 0.f};
        #pragma unroll 2
        for (int k0 = 0; k0 < K; k0 += 4) {
            v2f a, b;
            a.x = ap[0];
            a.y = ap[1];
            b.x = bp[0];
            b.y = bp[Ncols];
#if __has_builtin(__builtin_amdgcn_wmma_f32_16x16x4_f32)
            acc = __builtin_amdgcn_wmma_f32_16x16x4_f32(
                false, a, false, b, (short)0, acc, false, false);
#else
            for (int r = 0; r < 8; ++r) {
                int rr = r + 8 * hi;
                for (int kk = 0; kk < 4; ++kk)
                    acc[r] += As[rr * K + k0 + kk] * B[(size_t)(k0 + kk) * Ncols + col];
            }
#endif
            ap += 4;
            bp += (size_t)4 * Ncols;
        }
        // epilogue: bias hoisted, pointer-walked stores (no per-element branches)
        const float bv = bias ? bias[col] : 0.0f;
        float* cp = C + (size_t)(row0 + 8 * hi) * Ncols + col;  // VGPR r -> M = r + 8*hi
        if (act == 1) {
            #pragma unroll
            for (int r = 0; r < 8; ++r) { cp[0] = fmaxf(acc[r] + bv, 0.0f); cp += Ncols; }
        } else {
            #pragma unroll
            for (int r = 0; r < 8; ++r) { cp[0] = acc[r] + bv; cp += Ncols; }
        }
    }
}

// W1 (3 x 41 x 128) -> padded (3 x 44 x 128), zero rows 41..43
__global__ void k_pad_w1(const float* __restrict__ W1, float* __restrict__ Wp)
{
    int idx = blockIdx.x * 256 + threadIdx.x;
    if (idx >= 3 * KPAD * HDt) return;
    int i = idx / (KPAD * HDt), rem = idx - i * (KPAD * HDt);
    int k = rem / HDt, c = rem - k * HDt;
    Wp[idx] = (k < INt) ? W1[(size_t)i * INt * HDt + k * HDt + c] : 0.f;
}

// x0 padded: [N x 44] = [node_features(9) | type_emb[node_type](32) | 0,0,0]
__global__ void k_build_x0(const float* __restrict__ nf, const float* __restrict__ temb,
                           const int* __restrict__ ntype, float* __restrict__ x0)
{
    int idx = blockIdx.x * 256 + threadIdx.x;
    if (idx >= NN * KPAD) return;
    int n = idx / KPAD, c = idx - n * KPAD;
    float v;
    if (c < NFt)      v = nf[n * NFt + c];
    else if (c < INt) v = temb[ntype[n] * 32 + (c - NFt)];
    else              v = 0.f;
    x0[idx] = v;
}

// sm layout: [0..2] gates, [4..6] wts, [8..103] v1[i][h][d], [104..127] v2[i][d],
//            [128..151] selfvec[i][d] (gate * mean(ea))
__global__ void k_small_pre(const float* etg, const float* eta,
                            const float* We1, const float* ae1,
                            const float* We2, const float* ae2, float* sm)
{
    int t = threadIdx.x;
    if (t == 0) {
        sm[0] = 1.f / (1.f + expf(-etg[0]));
        sm[1] = 1.f / (1.f + expf(-etg[1]));
        sm[2] = 1.f / (1.f + expf(-etg[2]));
        float m  = fmaxf(eta[0], fmaxf(eta[1], eta[2]));
        float e0 = expf(eta[0] - m), e1 = expf(eta[1] - m), e2 = expf(eta[2] - m);
        float s  = e0 + e1 + e2;
        sm[4] = e0 / s; sm[5] = e1 / s; sm[6] = e2 / s;
    }
    if (t < 96) {  // v1[i][h][d] = sum_c We1[i][d][h*32+c] * att_e1[i][h][c]
        int i = t >> 5, rem = t & 31, h = rem >> 3, d = rem & 7;
        float s = 0.f;
        for (int c = 0; c < C1t; ++c)
            s += We1[i * EDt * HDt + d * HDt + h * C1t + c] * ae1[i * Ht * C1t + h * C1t + c];
        sm[8 + t] = s;
    }
    if (t >= 96 && t < 120) {  // v2[i][d] = sum_c We2[i][d][c] * att_e2[i][0][c]
        int j = t - 96, i = j >> 3, d = j & 7;
        float s = 0.f;
        for (int c = 0; c < HDt; ++c)
            s += We2[i * EDt * HDt + d * HDt + c] * ae2[i * HDt + c];
        sm[104 + j] = s;
    }
}

__global__ void k_mean_ea(const float* ea0, const float* ea1, const float* ea2, float* sm)
{
    __shared__ float red[256];
    __shared__ float tmp[8];
    int i = blockIdx.x, t = threadIdx.x;
    const float* ea = (i == 0) ? ea0 : (i == 1) ? ea1 : ea2;
    float loc[8] = {0, 0, 0, 0, 0, 0, 0, 0};
    for (int e = t; e < EE; e += 256)
        for (int d = 0; d < 8; ++d) loc[d] += ea[(size_t)e * 8 + d];
    for (int d = 0; d < 8; ++d) {
        red[t] = loc[d]; __syncthreads();
        for (int s = 128; s > 0; s >>= 1) { if (t < s) red[t] += red[t + s]; __syncthreads(); }
        if (t == 0) tmp[d] = red[0];
        __syncthreads();
    }
    if (t < 8) sm[128 + i * 8 + t] = sm[i] * tmp[t] / (float)EE;
}

// per-node attention scores + init of m/den/out accumulator
__global__ void k_node_scores(const float* __restrict__ h, const float* __restrict__ as,
                              const float* __restrict__ ad, float* ssrc, float* sdst,
                              float* mseg, float* den, float* out, int H, int C)
{
    int idx = blockIdx.x * blockDim.x + threadIdx.x;
    if (idx >= NN * H) return;
    int n = idx / H, hh = idx - n * H;
    const float* hv = h + (size_t)n * HDt + hh * C;
    const float* a1 = as + hh * C;
    const float* a2 = ad + hh * C;
    float s1 = 0.f, s2 = 0.f;
    for (int c = 0; c < C; ++c) { float x = hv[c]; s1 += x * a1[c]; s2 += x * a2[c]; }
    ssrc[idx] = s1; sdst[idx] = s2;
    mseg[idx] = -3e38f; den[idx] = 0.f;
    float* o = out + (size_t)n * HDt + hh * C;
    for (int c = 0; c < C; ++c) o[c] = 0.f;
}

__global__ void k_edge_alpha(const float* __restrict__ ea, const int* __restrict__ src,
                             const int* __restrict__ dst, const float* __restrict__ ssrc,
                             const float* __restrict__ sdst, const float* __restrict__ v,
                             const float* __restrict__ selfv, const float* __restrict__ gates,
                             int ti, float* alpha, float* mseg, int H)
{
    int e = blockIdx.x * blockDim.x + threadIdx.x;
    if (e >= EE + NN) return;
    float eav[8];
    int s, d;
    if (e < EE) {
        s = src[e]; d = dst[e];
        float g = gates[ti];
        for (int k = 0; k < 8; ++k) eav[k] = g * ea[(size_t)e * 8 + k];
    } else {
        s = d = e - EE;
        for (int k = 0; k < 8; ++k) eav[k] = selfv[k];
    }
    for (int hh = 0; hh < H; ++hh) {
        float sc = ssrc[s * H + hh] + sdst[d * H + hh];
        for (int k = 0; k < 8; ++k) sc += eav[k] * v[hh * 8 + k];
        float a = sc > 0.f ? sc : 0.2f * sc;   // leaky_relu(0.2)
        alpha[(size_t)e * H + hh] = a;
        atomicMaxF(&mseg[d * H + hh], a);
    }
}

__global__ void k_edge_exp(const int* __restrict__ dst, const float* __restrict__ mseg,
                           float* alpha, float* den, int H)
{
    int e = blockIdx.x * blockDim.x + threadIdx.x;
    if (e >= EE + NN) return;
    int d = (e < EE) ? dst[e] : e - EE;
    for (int hh = 0; hh < H; ++hh) {
        float a = expf(alpha[(size_t)e * H + hh] - mseg[d * H + hh]);
        alpha[(size_t)e * H + hh] = a;
        atomicAdd(&den[d * H + hh], a);
    }
}

// one wave per edge: out[dst] += h[src] * w  (128 channels)
__global__ __launch_bounds__(256)
void k_scatter(const float* __restrict__ h, const float* __restrict__ alpha,
               const float* __restrict__ den, const int* __restrict__ src,
               const int* __restrict__ dst, float* out, int H, int C)
{
    int nEdges = EE + NN;
    int wid  = (blockIdx.x * blockDim.x + threadIdx.x) >> 5;
    int lane = threadIdx.x & 31;
    int nw   = (gridDim.x * blockDim.x) >> 5;
    for (int e = wid; e < nEdges; e += nw) {
        int s, d;
        if (e < EE) { s = src[e]; d = dst[e]; } else { s = d = e - EE; }
        for (int c = lane; c < HDt; c += 32) {
            int hh = c / C;
            float w = alpha[(size_t)e * H + hh] / (den[d * H + hh] + 1e-16f);
            atomicAdd(&out[(size_t)d * HDt + c], h[(size_t)s * HDt + c] * w);
        }
    }
}

__global__ void k_finalize1(const float* __restrict__ out, const float* __restrict__ b,
                            float* __restrict__ h1)
{
    int idx = blockIdx.x * 256 + threadIdx.x;
    if (idx >= NN * HDt) return;
    float v = out[idx] + b[idx & (HDt - 1)];
    h1[idx] = v > 0.f ? v : expm1f(v);  // ELU
}

__global__ void k_finalize2(const float* __restrict__ out, const float* __restrict__ b,
                            const float* __restrict__ sm, int ti, float* __restrict__ acc)
{
    int idx = blockIdx.x * 256 + threadIdx.x;
    if (idx >= NN * HDt) return;
    float v = (out[idx] + b[idx & (HDt - 1)]) * sm[4 + ti];
    acc[idx] = (ti == 0) ? v : acc[idx] + v;
}

// scores[v] = veh[v] . pool_W + pool_b  (one wave per vehicle row)
__global__ void k_pool_scores(const float* __restrict__ acc, const float* __restrict__ pw,
                              const float* __restrict__ pb, float* __restrict__ scores)
{
    int wid  = (blockIdx.x * blockDim.x + threadIdx.x) >> 5;
    int lane = threadIdx.x & 31;
    int nw   = (gridDim.x * blockDim.x) >> 5;
    for (int v = wid; v < VVv; v += nw) {
        const float* row = acc + (size_t)(RRr + v) * HDt;
        float s = 0.f;
        for (int c = lane; c < HDt; c += 32) s += row[c] * pw[c];
        for (int off = 16; off > 0; off >>= 1) s += __shfl_xor(s, off, 32);
        if (lane == 0) scores[v] = s + pb[0];
    }
}

// masked softmax over 49936 cols + pooled = aw @ veh; writes comb[64 x 256]
__global__ __launch_bounds__(128)
void k_pool(const float* __restrict__ acc, const float* __restrict__ scores,
            const int* __restrict__ mask, float* __restrict__ comb)
{
    __shared__ float red[128];
    __shared__ float wl[128];
    int r = blockIdx.x, t = threadIdx.x;
    const int* mrow = mask + (size_t)r * VVv;

    float mx = -3e38f;
    for (int v = t; v < VVv; v += 128) if (mrow[v]) mx = fmaxf(mx, scores[v]);
    red[t] = mx; __syncthreads();
    for (int s = 64; s > 0; s >>= 1) { if (t < s) red[t] = fmaxf(red[t], red[t + s]); __syncthreads(); }
    float m = red[0];
    __syncthreads();

    float se = 0.f;
    for (int v = t; v < VVv; v += 128) if (mrow[v]) se += expf(scores[v] - m);
    red[t] = se; __syncthreads();
    for (int s = 64; s > 0; s >>= 1) { if (t < s) red[t] += red[t + s]; __syncthreads(); }
    float den = red[0];
    float inv = (m > -1e37f) ? 1.f / den : 0.f;   // no-mask row -> weights 0

    float pc = 0.f;
    for (int v0 = 0; v0 < VVv; v0 += 128) {
        int v = v0 + t;
        float w = 0.f;
        if (v < VVv && mrow[v]) w = expf(scores[v] - m) * inv;
        __syncthreads();
        wl[t] = w;
        __syncthreads();
        int lim = min(128, VVv - v0);
        for (int j = 0; j < lim; ++j) pc += wl[j] * acc[(size_t)(RRr + v0 + j) * HDt + t];
    }
    comb[(size_t)r * 256 + t]       = acc[(size_t)r * HDt + t];  // rsu half
    comb[(size_t)r * 256 + 128 + t] = pc;                        // pooled half
}

__global__ void k_tail(const float* eta, float* out)
{
    int t = threadIdx.x;
    if (t < 3) out[64 * 16 + t] = eta[t];
}

// ---------------------------------------------------------------------------
extern "C" void kernel_launch(void* const* d_in, const int* in_sizes, int n_in,
                              void* d_out, int out_size, void* d_ws, size_t ws_size,
                              hipStream_t stream)
{
    const float* nf    = (const float*)d_in[0];
    const float* ea_p[3] = {(const float*)d_in[1], (const float*)d_in[2], (const float*)d_in[3]};
    const float* temb  = (const float*)d_in[4];
    const float* etg   = (const float*)d_in[5];
    const float* eta   = (const float*)d_in[6];
    const float* W1    = (const float*)d_in[7];
    const float* as1   = (const float*)d_in[8];
    const float* ad1   = (const float*)d_in[9];
    const float* We1   = (const float*)d_in[10];
    const float* ae1   = (const float*)d_in[11];
    const float* b1    = (const float*)d_in[12];
    const float* W2    = (const float*)d_in[13];
    const float* as2   = (const float*)d_in[14];
    const float* ad2   = (const float*)d_in[15];
    const float* We2   = (const float*)d_in[16];
    const float* ae2   = (const float*)d_in[17];
    const float* b2    = (const float*)d_in[18];
    const float* poolW = (const float*)d_in[19];
    const float* poolB = (const float*)d_in[20];
    const float* Wo1   = (const float*)d_in[21];
    const float* bo1   = (const float*)d_in[22];
    const float* Wo2   = (const float*)d_in[23];
    const float* bo2   = (const float*)d_in[24];
    const float* Wo3   = (const float*)d_in[25];
    const float* bo3   = (const float*)d_in[26];
    const int*   ntype = (const int*)d_in[27];
    const int*   ei_p[3] = {(const int*)d_in[28], (const int*)d_in[29], (const int*)d_in[30]};
    const int*   cmask = (const int*)d_in[31];
    float*       qout  = (float*)d_out;

    float* W = (float*)d_ws;
    size_t o = 0;
    float* x0p   = W + o; o += (size_t)NN * KPAD;
    float* hbuf  = W + o; o += (size_t)NN * HDt;
    float* obuf  = W + o; o += (size_t)NN * HDt;
    float* h1buf = W + o; o += (size_t)NN * HDt;
    float* accb  = W + o; o += (size_t)NN * HDt;
    float* ssrc  = W + o; o += (size_t)NN * Ht;
    float* sdst  = W + o; o += (size_t)NN * Ht;
    float* mseg  = W + o; o += (size_t)NN * Ht;
    float* dseg  = W + o; o += (size_t)NN * Ht;
    float* albuf = W + o; o += (size_t)(EE + NN) * Ht;
    float* score = W + o; o += (size_t)VVv;
    float* comb  = W + o; o += 64 * 256;
    float* q1    = W + o; o += 64 * 128;
    float* q2    = W + o; o += 64 * 64;
    float* w1pad = W + o; o += (size_t)3 * KPAD * HDt;
    float* sm    = W + o; o += 256;

    const int EB = (EE + NN + 255) / 256;

    k_build_x0<<<(NN * KPAD + 255) / 256, 256, 0, stream>>>(nf, temb, ntype, x0p);
    k_pad_w1<<<(3 * KPAD * HDt + 255) / 256, 256, 0, stream>>>(W1, w1pad);
    k_small_pre<<<1, 128, 0, stream>>>(etg, eta, We1, ae1, We2, ae2, sm);
    k_mean_ea<<<3, 256, 0, stream>>>(ea_p[0], ea_p[1], ea_p[2], sm);

    for (int i = 0; i < 3; ++i) {
        const int* src = ei_p[i];
        const int* dst = ei_p[i] + EE;
        // ----- layer 1 (H=4, C=32) -----
        k_wmma_gemm<<<NN / 16, 128, 0, stream>>>(x0p, w1pad + (size_t)i * KPAD * HDt,
                                                 hbuf, nullptr, KPAD, HDt, 0);
        k_node_scores<<<(NN * Ht + 255) / 256, 256, 0, stream>>>(
            hbuf, as1 + i * HDt, ad1 + i * HDt, ssrc, sdst, mseg, dseg, obuf, Ht, C1t);
        k_edge_alpha<<<EB, 256, 0, stream>>>(ea_p[i], src, dst, ssrc, sdst,
                                             sm + 8 + i * 32, sm + 128 + i * 8, sm, i,
                                             albuf, mseg, Ht);
        k_edge_exp<<<EB, 256, 0, stream>>>(dst, mseg, albuf, dseg, Ht);
        k_scatter<<<4096, 256, 0, stream>>>(hbuf, albuf, dseg, src, dst, obuf, Ht, C1t);
        k_finalize1<<<(NN * HDt + 255) / 256, 256, 0, stream>>>(obuf, b1 + i * HDt, h1buf);
        // ----- layer 2 (H=1, C=128) -----
        k_wmma_gemm<<<NN / 16, 128, 0, stream>>>(h1buf, W2 + (size_t)i * HDt * HDt,
                                                 hbuf, nullptr, HDt, HDt, 0);
        k_node_scores<<<(NN + 255) / 256, 256, 0, stream>>>(
            hbuf, as2 + i * HDt, ad2 + i * HDt, ssrc, sdst, mseg, dseg, obuf, 1, HDt);
        k_edge_alpha<<<EB, 256, 0, stream>>>(ea_p[i], src, dst, ssrc, sdst,
                                             sm + 104 + i * 8, sm + 128 + i * 8, sm, i,
                                             albuf, mseg, 1);
        k_edge_exp<<<EB, 256, 0, stream>>>(dst, mseg, albuf, dseg, 1);
        k_scatter<<<4096, 256, 0, stream>>>(hbuf, albuf, dseg, src, dst, obuf, 1, HDt);
        k_finalize2<<<(NN * HDt + 255) / 256, 256, 0, stream>>>(obuf, b2 + i * HDt, sm, i, accb);
    }

    k_pool_scores<<<1024, 256, 0, stream>>>(accb, poolW, poolB, score);
    k_pool<<<64, 128, 0, stream>>>(accb, score, cmask, comb);

    // MLP head (WMMA): 64x256 @ 256x128 -> relu -> @128x64 -> relu -> @64x16
    k_wmma_gemm<<<4, 128, 0, stream>>>(comb, Wo1, q1, bo1, 256, 128, 1);
    k_wmma_gemm<<<4, 128, 0, stream>>>(q1, Wo2, q2, bo2, 128, 64, 1);
    k_wmma_gemm<<<4, 128, 0, stream>>>(q2, Wo3, qout, bo3, 64, 16, 0);

    k_tail<<<1, 32, 0, stream>>>(eta, qout);
}